// BlittingStrokeModel_48876727829124
// MI455X (gfx1250) — compile-verified
//
#include <hip/hip_runtime.h>
#include <math.h>

#define NSTK 32
#define HIMG 512
#define WIMG 512
#define HB   64

typedef float v2f __attribute__((ext_vector_type(2)));
typedef float v8f __attribute__((ext_vector_type(8)));
typedef unsigned int u32x4 __attribute__((ext_vector_type(4)));
typedef int  i32x4 __attribute__((ext_vector_type(4)));
typedef int  i32x8 __attribute__((ext_vector_type(8)));

#if defined(__has_builtin)
#if __has_builtin(__builtin_amdgcn_tensor_load_to_lds)
#define USE_TDM 1
#endif
#endif

struct __align__(16) StrokeP {
  float C0, C1, C2, D0;
  float D1, D2, colR, colG;
  float colB, colA, bx0, bx1;
  float by0, by1, act, pad;
};

// ---------------- Kernel 1: spline solve + per-stroke affine constants ----
__global__ void spline_kernel(const float* __restrict__ traj,
                              const float* __restrict__ colors,
                              StrokeP* __restrict__ params)
{
  const int b = threadIdx.x;
  if (b >= 4) return;
  const float* T  = traj + (size_t)b * 4 * NSTK;
  const float* tt = T;
  const float* qx = T + NSTK;
  const float* qy = T + 2 * NSTK;
  const float* qz = T + 3 * NSTK;

  float h[NSTK - 1], sx[NSTK - 1], sy[NSTK - 1];
  for (int i = 0; i < NSTK - 1; ++i) {
    h[i] = tt[i + 1] - tt[i];
    float inv = 1.0f / h[i];
    sx[i] = (qx[i + 1] - qx[i]) * inv;
    sy[i] = (qy[i + 1] - qy[i]) * inv;
  }
  // Thomas algorithm on interior nodes (natural BC: M[0]=M[N-1]=0), 2 RHS (x,y)
  float cp[NSTK], rx[NSTK], ry[NSTK];
  {
    float beta = 2.0f * (h[0] + h[1]);
    float invb = 1.0f / beta;
    cp[1] = h[1] * invb;
    rx[1] = 6.0f * (sx[1] - sx[0]) * invb;
    ry[1] = 6.0f * (sy[1] - sy[0]) * invb;
    for (int i = 2; i <= NSTK - 2; ++i) {
      float a = h[i - 1];
      beta = 2.0f * (h[i - 1] + h[i]) - a * cp[i - 1];
      invb = 1.0f / beta;
      cp[i] = h[i] * invb;
      rx[i] = (6.0f * (sx[i] - sx[i - 1]) - a * rx[i - 1]) * invb;
      ry[i] = (6.0f * (sy[i] - sy[i - 1]) - a * ry[i - 1]) * invb;
    }
  }
  float Mx[NSTK], My[NSTK];
  Mx[0] = My[0] = 0.0f;
  Mx[NSTK - 1] = My[NSTK - 1] = 0.0f;
  for (int i = NSTK - 2; i >= 1; --i) {
    Mx[i] = rx[i] - cp[i] * Mx[i + 1];
    My[i] = ry[i] - cp[i] * My[i + 1];
  }
  float ddx[NSTK], ddy[NSTK];
  const float sixth = 1.0f / 6.0f;
  for (int i = 0; i < NSTK - 1; ++i) {
    ddx[i] = sx[i] - h[i] * (2.0f * Mx[i] + Mx[i + 1]) * sixth;
    ddy[i] = sy[i] - h[i] * (2.0f * My[i] + My[i + 1]) * sixth;
  }
  ddx[NSTK - 1] = sx[NSTK - 2] + h[NSTK - 2] * (2.0f * Mx[NSTK - 1] + Mx[NSTK - 2]) * sixth;
  ddy[NSTK - 1] = sy[NSTK - 2] + h[NSTK - 2] * (2.0f * My[NSTK - 1] + My[NSTK - 2]) * sixth;

  const float cr = colors[b * 4 + 0], cg = colors[b * 4 + 1];
  const float cb = colors[b * 4 + 2], ca = colors[b * 4 + 3];
  const float ctr = 0.5f * (float)(HB - 1);  // 31.5

  for (int n = 0; n < NSTK; ++n) {
    float px = qx[n], py = qy[n], z = qz[n];
    float th  = -atan2f(ddy[n], ddx[n]);
    float cth = cosf(th), sth = sinf(th);
    float sc   = fminf(fmaxf(z, 0.001f), 1.0f);
    float invs = 1.0f / sc;
    StrokeP p;
    p.C1 = cth * invs;  p.C2 = -sth * invs;
    p.D1 = sth * invs;  p.D2 =  cth * invs;
    p.C0 = ctr - p.C1 * px - p.C2 * py;
    p.D0 = ctr - p.D1 * px - p.D2 * py;
    p.colR = cr; p.colG = cg; p.colB = cb; p.colA = ca;
    // footprint radius: |local| <= 32.5 maps back by scale*(|c|+|s|); +2 margin
    float r = sc * (fabsf(cth) + fabsf(sth)) * (ctr + 1.0f) + 2.0f;
    p.bx0 = px - r; p.bx1 = px + r;
    p.by0 = py - r; p.by1 = py + r;
    p.act = (z > 0.0f) ? 1.0f : 0.0f;
    p.pad = 0.0f;
    params[b * NSTK + n] = p;
  }
}

// ---- bilinear tap: unconditional LDS load from wrapped-safe index, masked --
__device__ __forceinline__ float tapA(const float* sA, int yi, int xi, float& m)
{
  bool inb = ((unsigned)yi < (unsigned)HB) & ((unsigned)xi < (unsigned)HB);
  m = inb ? 1.0f : 0.0f;
  int idx = ((yi & (HB - 1)) << 6) | (xi & (HB - 1));  // always a valid LDS index
  return sA[idx] * m;                                  // no divergent branch
}

// ---------------- Kernel 2: fused stroke compositing -----------------------
// One wave = one 16x16 pixel tile, matching the WMMA 16x16 f32 D layout:
//   lane L -> column (L & 15); VGPR j -> row j + 8*(L >> 4).
__global__ void __launch_bounds__(256)
composite_kernel(const float* __restrict__ imgIn,
                 const float* __restrict__ brush,
                 const StrokeP* __restrict__ params,
                 float* __restrict__ outp)
{
  __shared__ float sA[HB * HB];      // brush alpha plane, 16 KB

  const int b    = blockIdx.z;
  const int tid  = threadIdx.x;
  const int lane = tid & 31;
  const int wave = tid >> 5;

#ifdef USE_TDM
  if (wave == 0) {
    // Tensor DMA: 64x64 f32 tile of a 64x64 tensor -> LDS (CDNA5 TDM path)
    unsigned long long ga = (unsigned long long)(const void*)(brush + 3 * HB * HB);
    unsigned int lds = (unsigned int)(unsigned long long)(const void*)&sA[0];
    u32x4 g0;
    g0[0] = 1u;                                                   // count=1, user mode
    g0[1] = lds;                                                  // lds_addr
    g0[2] = (unsigned int)ga;                                     // global_addr[31:0]
    g0[3] = ((unsigned int)(ga >> 32) & 0x01FFFFFFu) | 0x80000000u; // addr[56:32] | type=2
    i32x8 g1;
    g1[0] = (int)(2u << 16);                    // data_size = 4 bytes; wg_mask = 0
    g1[1] = (int)((unsigned)HB << 16);          // tensor_dim0 = 64
    g1[2] = (int)((unsigned)HB << 16);          // tensor_dim1 = 64
    g1[3] = (int)((unsigned)HB << 16);          // tile_dim0 = 64
    g1[4] = (int)(unsigned)HB;                  // tile_dim1 = 64
    g1[5] = HB;                                 // tensor_dim0_stride = 64
    g1[6] = 0;
    g1[7] = 0;
    i32x4 gz  = {0, 0, 0, 0};                   // groups 2/3 unused (2D tensor)
    i32x8 gz8 = {0, 0, 0, 0, 0, 0, 0, 0};       // extra descriptor group (unused)
    __builtin_amdgcn_tensor_load_to_lds(g0, g1, gz, gz, gz8, 0);
    __builtin_amdgcn_s_wait_tensorcnt(0);
  }
#else
  {
    const float* ba = brush + 3 * HB * HB;
    for (int i = tid; i < HB * HB; i += 256) sA[i] = ba[i];
  }
#endif
  __syncthreads();

  // tile origin: block covers 64x32 px; 8 waves in a 4x2 tile arrangement
  const int tx0 = blockIdx.x * 64 + (wave & 3) * 16;
  const int ty0 = blockIdx.y * 32 + (wave >> 2) * 16;
  const int col  = lane & 15;
  const int half = lane >> 4;
  const int xPix  = tx0 + col;
  const int yBase = ty0 + half * 8;

  const size_t base  = (size_t)b * 4 * HIMG * WIMG;
  const size_t plane = (size_t)HIMG * WIMG;

  // load image, init oil-space accumulators (1 - rgb)
  float accR[8], accG[8], accB[8];
#pragma unroll
  for (int j = 0; j < 8; ++j) {
    size_t off = (size_t)(yBase + j) * WIMG + xPix;
    accR[j] = 1.0f - imgIn[base + 0 * plane + off];
    accG[j] = 1.0f - imgIn[base + 1 * plane + off];
    accB[j] = 1.0f - imgIn[base + 2 * plane + off];
  }

  // WMMA A operand (stroke-invariant): A[M,:] = [1, M, 0, 0]
  // 32-bit A 16x4 layout: lanes 0-15 hold K=0,1 (M=lane); lanes 16-31 hold K=2,3 (=0)
  v2f amat;
  amat[0] = half ? 0.0f : 1.0f;
  amat[1] = half ? 0.0f : (float)lane;
  const float nf = (float)col;
  const v8f czero = {};

  const StrokeP* __restrict__ pb = params + b * NSTK;  // uniform base -> SGPR params

  for (int n = 0; n < NSTK; ++n) {
    const StrokeP p = pb[n];   // uniform address: scalarizes to s_load
    // wave-uniform cull; force scalar branch so EXEC stays all-ones for WMMA
    int skip = (p.act == 0.0f) |
               ((float)(tx0 + 15) < p.bx0) | ((float)tx0 > p.bx1) |
               ((float)(ty0 + 15) < p.by0) | ((float)ty0 > p.by1);
    if (__builtin_amdgcn_readfirstlane(skip)) continue;

    const float e0 = p.C0 + p.C1 * (float)tx0 + p.C2 * (float)ty0;
    const float f0 = p.D0 + p.D1 * (float)tx0 + p.D2 * (float)ty0;

    // B operand: B[0,N] = e0 + C1*N ; B[1,N] = C2 ; rows 2,3 = 0
    // 32-bit B 4x16 layout: lanes 0-15 hold K=0,1 ; lanes 16-31 hold K=2,3
    v2f bxm, bym;
    bxm[0] = half ? 0.0f : (e0 + p.C1 * nf);
    bxm[1] = half ? 0.0f : p.C2;
    bym[0] = half ? 0.0f : (f0 + p.D1 * nf);
    bym[1] = half ? 0.0f : p.D2;

    // 256 (lx,ly) coordinate pairs for the tile on the matrix pipe
    v8f lx = __builtin_amdgcn_wmma_f32_16x16x4_f32(false, amat, false, bxm,
                                                   (short)0, czero, false, false);
    v8f ly = __builtin_amdgcn_wmma_f32_16x16x4_f32(false, amat, false, bym,
                                                   (short)0, czero, false, false);

#pragma unroll
    for (int j = 0; j < 8; ++j) {
      float X = lx[j], Y = ly[j];
      float xf = floorf(X), yf = floorf(Y);
      float wx = X - xf,  wy = Y - yf;
      int  xi = (int)xf,  yi = (int)yf;

      float m00, m01, m10, m11;
      float a00 = tapA(sA, yi,     xi,     m00);
      float a01 = tapA(sA, yi,     xi + 1, m01);
      float a10 = tapA(sA, yi + 1, xi,     m10);
      float a11 = tapA(sA, yi + 1, xi + 1, m11);

      float at = a00 + (a01 - a00) * wx;
      float ab = a10 + (a11 - a10) * wx;
      float asmp = at + (ab - at) * wy;            // bilinear brush alpha
      float mt = m00 + (m01 - m00) * wx;
      float mb = m10 + (m11 - m10) * wx;
      float wc = mt + (mb - mt) * wy;              // coverage weight

      float av = asmp * p.colA;
      float om = 1.0f - av;
      accR[j] = accR[j] * om + (1.0f - p.colR * wc) * av;
      accG[j] = accG[j] * om + (1.0f - p.colG * wc) * av;
      accB[j] = accB[j] * om + (1.0f - p.colB * wc) * av;
    }
  }

  // write back: rgb = 1 - acc; alpha passthrough
#pragma unroll
  for (int j = 0; j < 8; ++j) {
    size_t off = (size_t)(yBase + j) * WIMG + xPix;
    outp[base + 0 * plane + off] = 1.0f - accR[j];
    outp[base + 1 * plane + off] = 1.0f - accG[j];
    outp[base + 2 * plane + off] = 1.0f - accB[j];
    outp[base + 3 * plane + off] = imgIn[base + 3 * plane + off];
  }
}

// ---------------- launcher -------------------------------------------------
extern "C" void kernel_launch(void* const* d_in, const int* in_sizes, int n_in,
                              void* d_out, int out_size, void* d_ws, size_t ws_size,
                              hipStream_t stream)
{
  const float* images = (const float*)d_in[0];  // (4,4,512,512)
  const float* traj   = (const float*)d_in[1];  // (4,4,32)
  const float* colors = (const float*)d_in[2];  // (4,4)
  const float* brush  = (const float*)d_in[3];  // (4,64,64)
  float* outp = (float*)d_out;
  StrokeP* params = (StrokeP*)d_ws;             // 4*32*64 B = 8 KB

  spline_kernel<<<dim3(1), dim3(4), 0, stream>>>(traj, colors, params);

  dim3 grid(WIMG / 64, HIMG / 32, 4);
  composite_kernel<<<grid, 256, 0, stream>>>(images, brush, params, outp);
}